// TensorProductContractionLayer_26079041421544
// MI455X (gfx1250) — compile-verified
//
#include <hip/hip_runtime.h>

// CGC tensor-product contraction: res[row, h] = sum_mn cgc[mn, h] * x[row,m] * y[row,n]
// row = i*128+f (12.8M rows), mn = m*3+n in [0,9) padded to 12, h in [0,5) padded to 16.
// Mapped onto V_WMMA_F32_16X16X4_F32: 16-row tiles, 3 K-steps of 4.
//
// Memory-bound problem (563 MB @ 23.3 TB/s ~ 24 us floor); WMMA offloads the
// 45-FMA/row contraction so the wave is dominated by coalesced b96 streaming.

typedef float v2f __attribute__((ext_vector_type(2)));
typedef float v8f __attribute__((ext_vector_type(8)));

struct F3 { float a, b, c; };   // one 12-byte row of x or y (coalesced b96 loads)

__global__ __launch_bounds__(256) void cgc_tp_wmma_kernel(
    const float* __restrict__ xg, const float* __restrict__ yg,
    const float* __restrict__ cgc, float* __restrict__ out,
    long long nrows)
{
    const int  lane = threadIdx.x & 31;
    const int  col  = lane & 15;        // N column of B/D; also M row this lane co-owns in A
    const bool hi   = lane >= 16;       // lane half: holds K pair {2,3} of each WMMA step

    const long long wavesPerBlock = blockDim.x >> 5;
    const long long waveId     = (long long)blockIdx.x * wavesPerBlock + (threadIdx.x >> 5);
    const long long totalWaves = (long long)gridDim.x * wavesPerBlock;
    const long long stride     = totalWaves * 16;

    // ---- Build B operand once per wave: cgc padded to 12x16 (K x N). ----
    // Step kk, VGPR pair: lo lanes hold K rows {4kk, 4kk+1}; hi lanes {4kk+2, 4kk+3}.
    v2f b[3];
#pragma unroll
    for (int kk = 0; kk < 3; ++kk) {
        const int mn0 = 4 * kk + (hi ? 2 : 0);
        float b0 = 0.0f, b1 = 0.0f;
        if (col < 5) {
            if (mn0     < 9) b0 = cgc[mn0 * 5 + col];
            if (mn0 + 1 < 9) b1 = cgc[(mn0 + 1) * 5 + col];
        }
        b[kk].x = b0;
        b[kk].y = b1;
    }

    const F3* __restrict__ X = reinterpret_cast<const F3*>(xg);
    const F3* __restrict__ Y = reinterpret_cast<const F3*>(yg);

    // nrows is a multiple of 16 (12,800,000), so every started tile is full.
    for (long long base = waveId * 16; base < nrows; base += stride) {
        const long long row = base + col;   // lanes L and L+16 share row M = col

        F3 xv = X[row];
        F3 yv = Y[row];

        // Prefetch next tile (gfx1250 global_prefetch_b8; speculative, OOB-safe).
        {
            long long nrow = row + stride;
            if (nrow >= nrows) nrow = row;
            __builtin_prefetch(&X[nrow], 0, 1);
            __builtin_prefetch(&Y[nrow], 0, 1);
        }

        // ---- A operand: outer products p[k] = x[k/3]*y[k%3], K padded 9->12. ----
        // lo lane: k = {0,1}, {4,5}, {8,pad} ; hi lane: k = {2,3}, {6,7}, {pad,pad}
        v2f a0, a1, a2;
        a0.x = xv.a * (hi ? yv.c : yv.a);               // k=2 | k=0
        a0.y = (hi ? xv.b : xv.a) * (hi ? yv.a : yv.b); // k=3 | k=1
        a1.x = (hi ? xv.c : xv.b) * (hi ? yv.a : yv.b); // k=6 | k=4
        a1.y = (hi ? xv.c : xv.b) * (hi ? yv.b : yv.c); // k=7 | k=5
        a2.x = hi ? 0.0f : xv.c * yv.c;                 // pad | k=8
        a2.y = 0.0f;                                    // pad | pad

        // ---- D = A x B over 3 K-steps (EXEC all-ones here, per WMMA rules). ----
        v8f c = {};
        c = __builtin_amdgcn_wmma_f32_16x16x4_f32(false, a0, false, b[0],
                                                  (short)0, c, false, false);
        c = __builtin_amdgcn_wmma_f32_16x16x4_f32(false, a1, false, b[1],
                                                  (short)0, c, false, false);
        c = __builtin_amdgcn_wmma_f32_16x16x4_f32(false, a2, false, b[2],
                                                  (short)0, c, false, false);

        // ---- Store: D VGPR r holds (M = r + 8*hi, N = col); keep N < 5. ----
        // Wave writes the contiguous 320B block out[base*5 .. (base+16)*5).
        if (col < 5) {
#pragma unroll
            for (int r = 0; r < 8; ++r) {
                const long long m = base + r + (hi ? 8 : 0);
                out[m * 5 + col] = c[r];
            }
        }
    }
}

extern "C" void kernel_launch(void* const* d_in, const int* in_sizes, int n_in,
                              void* d_out, int out_size, void* d_ws, size_t ws_size,
                              hipStream_t stream) {
    const float* x   = (const float*)d_in[0];  // (N_ATOMS, 128, 3) f32
    const float* y   = (const float*)d_in[1];  // (N_ATOMS, 128, 3) f32
    const float* cgc = (const float*)d_in[2];  // (3, 3, 5) f32
    float* out = (float*)d_out;                // (N_ATOMS, 128, 5) f32

    const long long nrows = (long long)in_sizes[0] / 3;       // 12,800,000
    const long long tiles = (nrows + 15) / 16;                // 16-row WMMA tiles
    long long blocks = (tiles + 7) / 8;                       // 8 waves per block
    if (blocks > 32768) blocks = 32768;                       // grid-stride beyond this
    if (blocks < 1) blocks = 1;

    cgc_tp_wmma_kernel<<<(int)blocks, 256, 0, stream>>>(x, y, cgc, out, nrows);
}